// Model_64896955843173
// MI455X (gfx1250) — compile-verified
//
#include <hip/hip_runtime.h>
#include <math.h>

// ---------------------------------------------------------------------------
// Model constants
// ---------------------------------------------------------------------------
#define BATCH     2
#define SEQ_LEN   512
#define ENC_IN    64
#define PRED_LEN  96
#define PATCH_LEN 16
#define STRIDE_   8
#define D_MODEL   256
#define D_MODEL1  64
#define K_TIME    64
#define K_CH      32
#define STATE     32
#define PATCH_NUM 63        // (512-16)/8+1
#define NROWS     128       // BATCH*ENC_IN

typedef float v2f __attribute__((ext_vector_type(2)));
typedef float v8f __attribute__((ext_vector_type(8)));

// ---------------------------------------------------------------------------
// Generic fp32 WMMA GEMM:  C[M,N] = act( A[M,K] @ B[K,N] + bias (+ Cold) )
// One wave computes one 16x16 C tile with V_WMMA_F32_16X16X4_F32.
// The k-loop is fully branch-free: OOB lanes are clamped to a valid row/col
// (their garbage lands only in C elements the epilogue masks off), and the
// K%4 tail uses value-selects on unconditional in-bounds loads.
// act: 0=none, 1=softplus, 2=silu
// ---------------------------------------------------------------------------
__global__ __launch_bounds__(256)
void wmma_gemm_f32(const float* __restrict__ A, int lda,
                   const float* __restrict__ B, int ldb,
                   float* __restrict__ C, int ldc,
                   int M, int N, int K,
                   const float* __restrict__ bias,
                   int act, int accum)
{
    const int Mt = (M + 15) >> 4;
    const int Nt = (N + 15) >> 4;
    const int tile = blockIdx.x * (blockDim.x >> 5) + (threadIdx.x >> 5);
    if (tile >= Mt * Nt) return;               // wave-uniform exit
    const int mt = tile / Nt, nt = tile % Nt;
    const int lane = threadIdx.x & 31;
    const int half = lane >> 4;                // 0: K0/K1, 1: K2/K3 (and rows M+8)
    const int l15  = lane & 15;
    const int koff = half << 1;
    const int mbase = mt << 4, nbase = nt << 4;
    const int m = mbase + l15;                 // A row for this lane
    const int n = nbase + l15;                 // B/C column for this lane

    // clamp OOB lanes to a valid address: no predication needed in the k-loop
    const int mA = (m < M) ? m : (M - 1);
    const int nB = (n < N) ? n : (N - 1);
    const float* __restrict__ Ap = A + (size_t)mA * lda + koff;
    const float* __restrict__ Bp = B + (size_t)koff * ldb + nB;

    // uniform-condition prefetch of distant weight rows (gfx1250 global_prefetch)
    if (K > 64) __builtin_prefetch(&Bp[(size_t)64 * ldb], 0, 1);

    v8f c = {0.f,0.f,0.f,0.f,0.f,0.f,0.f,0.f};

    const int Kmain = K & ~3;
    #pragma unroll 4
    for (int k = 0; k < Kmain; k += 4) {
        v2f a, b;
        a.x = Ap[k];
        a.y = Ap[k + 1];
        b.x = Bp[(size_t)k * ldb];
        b.y = Bp[(size_t)(k + 1) * ldb];
        c = __builtin_amdgcn_wmma_f32_16x16x4_f32(
                /*neg_a=*/false, a, /*neg_b=*/false, b,
                /*c_mod=*/(short)0, c, /*reuse_a=*/false, /*reuse_b=*/false);
    }
    if (Kmain < K) {                           // K % 4 tail (branch-free masking)
        const int k0 = Kmain + koff, k1 = k0 + 1;
        const int k0c = (k0 < K) ? k0 : (K - 1);
        const int k1c = (k1 < K) ? k1 : (K - 1);
        const float a0 = A[(size_t)mA * lda + k0c];
        const float a1 = A[(size_t)mA * lda + k1c];
        const float b0 = B[(size_t)k0c * ldb + nB];
        const float b1 = B[(size_t)k1c * ldb + nB];
        v2f a, b;
        a.x = (k0 < K) ? a0 : 0.f;
        a.y = (k1 < K) ? a1 : 0.f;
        b.x = (k0 < K) ? b0 : 0.f;
        b.y = (k1 < K) ? b1 : 0.f;
        c = __builtin_amdgcn_wmma_f32_16x16x4_f32(
                false, a, false, b, (short)0, c, false, false);
    }

    if (n < N) {
        const float bv = bias ? bias[n] : 0.f;
        #pragma unroll
        for (int v = 0; v < 8; ++v) {
            const int row = mbase + v + (half << 3);
            if (row < M) {
                float val = c[v] + bv;
                if (accum) val += C[(size_t)row * ldc + n];
                if (act == 1)       val = (val > 20.f) ? val : log1pf(expf(val));
                else if (act == 2)  val = val / (1.f + expf(-val));
                C[(size_t)row * ldc + n] = val;
            }
        }
    }
}

// ---------------------------------------------------------------------------
// RevIN normalize: x (B,T,C) -> z (B*C, T); stores per-(b,c) mean/std
// ---------------------------------------------------------------------------
__global__ __launch_bounds__(256)
void revin_norm_kernel(const float* __restrict__ x,
                       const float* __restrict__ rw, const float* __restrict__ rb,
                       float* __restrict__ z,
                       float* __restrict__ mean, float* __restrict__ stdv)
{
    const int bc = blockIdx.x;                 // 0..127
    const int b = bc >> 6, c = bc & 63;
    const int t = threadIdx.x;
    __shared__ float s1[256], s2[256];
    float s = 0.f, q = 0.f;
    for (int tt = t; tt < SEQ_LEN; tt += 256) {
        float v = x[((size_t)b * SEQ_LEN + tt) * ENC_IN + c];
        s += v; q += v * v;
    }
    s1[t] = s; s2[t] = q; __syncthreads();
    for (int off = 128; off > 0; off >>= 1) {
        if (t < off) { s1[t] += s1[t + off]; s2[t] += s2[t + off]; }
        __syncthreads();
    }
    const float mu = s1[0] * (1.f / SEQ_LEN);
    const float var = s2[0] * (1.f / SEQ_LEN) - mu * mu;
    const float sd = sqrtf(var + 1e-5f);
    if (t == 0) { mean[bc] = mu; stdv[bc] = sd; }
    const float w = rw[c], bb = rb[c];
    for (int tt = t; tt < SEQ_LEN; tt += 256) {
        float v = x[((size_t)b * SEQ_LEN + tt) * ENC_IN + c];
        z[(size_t)bc * SEQ_LEN + tt] = (v - mu) / sd * w + bb;
    }
}

// ---------------------------------------------------------------------------
// Patch embed + transpose:  zt[bc][j][p] = sum_i z[bc][p*8+i]*W[i][j] + b[j]
// zt stored (128, 64, 64) with p-stride 64 (p=63 column unused).
// ---------------------------------------------------------------------------
__global__ __launch_bounds__(256)
void patch_embed_kernel(const float* __restrict__ z,
                        const float* __restrict__ w,   // (16,64)
                        const float* __restrict__ b,   // (64)
                        float* __restrict__ zt)
{
    const int row = blockIdx.x;                // bc
    __shared__ float zs[SEQ_LEN];
    __shared__ float ws[PATCH_LEN * D_MODEL1];
    for (int i = threadIdx.x; i < SEQ_LEN; i += blockDim.x) zs[i] = z[(size_t)row * SEQ_LEN + i];
    for (int i = threadIdx.x; i < PATCH_LEN * D_MODEL1; i += blockDim.x) ws[i] = w[i];
    __syncthreads();
    for (int e = threadIdx.x; e < D_MODEL1 * PATCH_NUM; e += blockDim.x) {
        const int j = e / PATCH_NUM, p = e % PATCH_NUM;
        float acc = b[j];
        #pragma unroll
        for (int i = 0; i < PATCH_LEN; ++i)
            acc += zs[p * STRIDE_ + i] * ws[i * D_MODEL1 + j];
        zt[((size_t)row * 64 + j) * 64 + p] = acc;
    }
}

// ---------------------------------------------------------------------------
// Depthwise causal (dir=0) / anti-causal reversed-tap (dir=1) conv + SiLU.
// xz rows are tokens (nrows*64, ld=ldxz); xi part is cols [0,din).
// ---------------------------------------------------------------------------
__global__ __launch_bounds__(256)
void conv_silu_kernel(const float* __restrict__ xz, int ldxz,
                      const float* __restrict__ cw, const float* __restrict__ cb,
                      float* __restrict__ xi, int ldxi,
                      int din, int dir)
{
    const int row = blockIdx.x;
    for (int e = threadIdx.x; e < 64 * din; e += blockDim.x) {
        const int l = e / din, ch = e % din;
        float acc = cb[ch];
        if (dir == 0) {
            #pragma unroll
            for (int k = 0; k < 4; ++k) {
                const int ls = l - 3 + k;
                if (ls >= 0) acc += xz[((size_t)row * 64 + ls) * ldxz + ch] * cw[ch * 4 + k];
            }
        } else {  // flipped-sequence mamba expressed in original orientation
            #pragma unroll
            for (int mtap = 0; mtap < 4; ++mtap) {
                const int ls = l + mtap;
                if (ls < 64) acc += xz[((size_t)row * 64 + ls) * ldxz + ch] * cw[ch * 4 + (3 - mtap)];
            }
        }
        xi[((size_t)row * 64 + l) * ldxi + ch] = acc / (1.f + expf(-acc));
    }
}

// ---------------------------------------------------------------------------
// Selective scan: lane ch owns h[32]; B/C rows staged through LDS per step.
// Fuses  y = (scan + xi*D) * silu(res)  in-register.
// ---------------------------------------------------------------------------
__global__ void mamba_scan_kernel(const float* __restrict__ delta, int lddl,
                                  const float* __restrict__ xi, int ldxi,
                                  const float* __restrict__ xdbl, int ldxd, int dtr,
                                  const float* __restrict__ xz, int ldxz,
                                  const float* __restrict__ A_log,
                                  const float* __restrict__ Dp,
                                  float* __restrict__ y, int ldy,
                                  int din, int dir)
{
    const int row = blockIdx.x;
    const int ch = threadIdx.x;
    __shared__ float BC[2 * STATE];            // Bm | Cm for the current token
    float A[STATE], h[STATE];
    float Dv = 0.f;
    if (ch < din) {
        #pragma unroll
        for (int n = 0; n < STATE; ++n) { A[n] = -expf(A_log[(size_t)ch * STATE + n]); h[n] = 0.f; }
        Dv = Dp[ch];
    }
    for (int step = 0; step < 64; ++step) {
        const int l = dir ? (63 - step) : step;
        const size_t tok = (size_t)row * 64 + l;
        __syncthreads();
        for (int i = threadIdx.x; i < 2 * STATE; i += blockDim.x)
            BC[i] = xdbl[tok * ldxd + dtr + i];
        __syncthreads();
        if (ch < din) {
            const float dlt = delta[tok * lddl + ch];
            const float xv  = xi[tok * ldxi + ch];
            const float du  = dlt * xv;
            float acc = 0.f;
            #pragma unroll
            for (int n = 0; n < STATE; ++n) {
                const float hn = expf(dlt * A[n]) * h[n] + du * BC[n];
                h[n] = hn;
                acc += hn * BC[STATE + n];
            }
            float yv = acc + xv * Dv;
            const float r = xz[tok * ldxz + din + ch];
            yv *= r / (1.f + expf(-r));
            y[tok * ldy + ch] = yv;
        }
    }
}

// ---------------------------------------------------------------------------
// Row softmax (in-place safe)
// ---------------------------------------------------------------------------
__global__ __launch_bounds__(256)
void softmax_rows_kernel(const float* __restrict__ in, float* __restrict__ out, int width)
{
    const int row = blockIdx.x;
    const int t = threadIdx.x;
    __shared__ float red[256];
    float mx = -1e30f;
    for (int i = t; i < width; i += 256) mx = fmaxf(mx, in[(size_t)row * width + i]);
    red[t] = mx; __syncthreads();
    for (int off = 128; off > 0; off >>= 1) { if (t < off) red[t] = fmaxf(red[t], red[t + off]); __syncthreads(); }
    mx = red[0]; __syncthreads();
    float s = 0.f;
    for (int i = t; i < width; i += 256) s += expf(in[(size_t)row * width + i] - mx);
    red[t] = s; __syncthreads();
    for (int off = 128; off > 0; off >>= 1) { if (t < off) red[t] += red[t + off]; __syncthreads(); }
    s = red[0];
    const float inv = 1.f / s;
    for (int i = t; i < width; i += 256)
        out[(size_t)row * width + i] = expf(in[(size_t)row * width + i] - mx) * inv;
}

// ---------------------------------------------------------------------------
// RevIN denorm + transpose: src (128, len) [row=b*64+c] -> dst (2, len, 64)
// ---------------------------------------------------------------------------
__global__ __launch_bounds__(256)
void denorm_kernel(const float* __restrict__ src, int len,
                   const float* __restrict__ rw, const float* __restrict__ rb,
                   const float* __restrict__ mean, const float* __restrict__ stdv,
                   float* __restrict__ dst)
{
    const size_t idx = (size_t)blockIdx.x * blockDim.x + threadIdx.x;
    const size_t total = (size_t)BATCH * len * ENC_IN;
    if (idx >= total) return;
    const int c = (int)(idx % ENC_IN);
    const int t = (int)((idx / ENC_IN) % len);
    const int b = (int)(idx / ((size_t)ENC_IN * len));
    const int bc = b * ENC_IN + c;
    const float v = src[(size_t)bc * len + t];
    dst[idx] = (v - rb[c]) / (rw[c] + 1e-10f) * stdv[bc] + mean[bc];
}

// ---------------------------------------------------------------------------
// Host-side helpers
// ---------------------------------------------------------------------------
static void gemm(hipStream_t s, const float* A, int lda, const float* B, int ldb,
                 float* C, int ldc, int M, int N, int K,
                 const float* bias, int act, int accum)
{
    const int Mt = (M + 15) >> 4, Nt = (N + 15) >> 4;
    const long tiles = (long)Mt * Nt;
    const int wpb = 8;                         // 8 waves (tiles) per 256-thread block
    const int blocks = (int)((tiles + wpb - 1) / wpb);
    wmma_gemm_f32<<<blocks, wpb * 32, 0, s>>>(A, lda, B, ldb, C, ldc, M, N, K, bias, act, accum);
}

struct MambaP {
    const float *in_w, *in_b, *cw, *cb, *xproj, *dtw, *dtb, *Alog, *Dp, *ow;
    int d, din, dtr;
};

static void run_mamba(hipStream_t s, const MambaP& p, int nrows,
                      const float* Xm, int ldx, float* outC, int ldout, int accum, int dir,
                      float* xz, int ldxz, float* xi, int ldxi, float* xdbl,
                      float* delta, int lddl, float* ybuf, int ldy)
{
    const int M = nrows * 64;
    // in_proj: xz = X @ in_w + in_b                     (WMMA)
    gemm(s, Xm, ldx, p.in_w, 2 * p.din, xz, ldxz, M, 2 * p.din, p.d, p.in_b, 0, 0);
    // depthwise conv (+dir) + SiLU
    conv_silu_kernel<<<nrows, 256, 0, s>>>(xz, ldxz, p.cw, p.cb, xi, ldxi, p.din, dir);
    // x_proj: xdbl = xi @ xproj_w                       (WMMA)
    gemm(s, xi, ldxi, p.xproj, p.dtr + 2 * STATE, xdbl, 80, M, p.dtr + 2 * STATE, p.din, nullptr, 0, 0);
    // delta = softplus(dt @ dt_w + dt_b)                (WMMA, fused act)
    gemm(s, xdbl, 80, p.dtw, p.din, delta, lddl, M, p.din, p.dtr, p.dtb, 1, 0);
    // selective scan (+xi*D, *silu(res))
    const int bd = (p.din + 127) & ~127;       // 128 (patch) or 512 (channel)
    mamba_scan_kernel<<<nrows, bd, 0, s>>>(delta, lddl, xi, ldxi, xdbl, 80, p.dtr,
                                           xz, ldxz, p.Alog, p.Dp, ybuf, ldy, p.din, dir);
    // out_proj: out (+)= y @ out_w                      (WMMA, optional accum)
    gemm(s, ybuf, ldy, p.ow, p.d, outC, ldout, M, p.d, p.din, nullptr, 0, accum);
}

// ---------------------------------------------------------------------------
// kernel_launch
// ---------------------------------------------------------------------------
extern "C" void kernel_launch(void* const* d_in, const int* in_sizes, int n_in,
                              void* d_out, int out_size, void* d_ws, size_t ws_size,
                              hipStream_t stream)
{
    auto F = [&](int i) { return (const float*)d_in[i]; };
    const float* x_in   = F(0);
    const float* ld1_w  = F(1);  const float* ld1_b = F(2);
    const float* ltp_w  = F(3);  const float* ltp_b = F(4);
    const float* ldm_w  = F(5);  const float* ldm_b = F(6);
    const float* lch_w  = F(7);  const float* lch_b = F(8);
    const float* wz_w   = F(9);  const float* wz_b  = F(10);
    const float* wr_w   = F(11); const float* wr_b  = F(12);
    const float* E_tp   = F(13); const float* E_tpf = F(14);
    const float* E_ch   = F(15); const float* E_chf = F(16);
    const float* rv_w   = F(17); const float* rv_b  = F(18);

    auto mp = [&](int base, int d, int din, int dtr) {
        MambaP p; p.in_w = F(base); p.in_b = F(base + 1); p.cw = F(base + 2); p.cb = F(base + 3);
        p.xproj = F(base + 4); p.dtw = F(base + 5); p.dtb = F(base + 6);
        p.Alog = F(base + 7); p.Dp = F(base + 8); p.ow = F(base + 9);
        p.d = d; p.din = din; p.dtr = dtr; return p;
    };
    const MambaP m_patch = mp(19, 63, 126, 4);
    const MambaP m_chf   = mp(29, 256, 512, 16);
    const MambaP m_chb   = mp(39, 256, 512, 16);

    // ---- workspace carve-up (floats) ----
    float* ws = (float*)d_ws;
    size_t o = 0;
    auto take = [&](size_t n) { float* p = ws + o; o += n; return p; };
    float* mean   = take(128);
    float* stdv   = take(128);
    float* z      = take((size_t)NROWS * SEQ_LEN);       // (128,512)
    float* zt_a   = take((size_t)NROWS * 64 * 64);       // (8192,64) ld64
    float* zt_b   = take((size_t)NROWS * 64 * 64);
    float* ztf    = take((size_t)NROWS * 64 * 63);       // (8192,63) ld63 -> rows of 4032
    float* xz     = take((size_t)8192 * 252);            // shared mamba scratch (max use)
    float* xi     = take((size_t)8192 * 128);
    float* xdbl   = take((size_t)8192 * 80);
    float* delta  = take((size_t)8192 * 128);
    float* ybuf   = take((size_t)8192 * 128);
    float* zA     = take((size_t)NROWS * 64);
    float* t_res  = take((size_t)NROWS * SEQ_LEN);
    float* z_time = take((size_t)NROWS * PRED_LEN);
    float* zc     = take((size_t)NROWS * D_MODEL);
    float* zc2    = take((size_t)NROWS * D_MODEL);
    float* zcA    = take((size_t)NROWS * K_CH);
    float* Esm    = take((size_t)K_CH * SEQ_LEN);
    float* EsmF   = take((size_t)K_CH * PRED_LEN);
    float* ch_res = take((size_t)NROWS * SEQ_LEN);
    float* zch    = take((size_t)NROWS * PRED_LEN);
    float* zo     = take((size_t)NROWS * PRED_LEN);
    float* trb    = take((size_t)NROWS * SEQ_LEN);
    (void)ws_size; (void)in_sizes; (void)n_in; (void)out_size;

    float* out = (float*)d_out;
    float* out_zout = out;                                     // (2,96,64)
    float* out_tr   = out + (size_t)BATCH * PRED_LEN * ENC_IN; // (2,512,64)
    float* out_cr   = out_tr + (size_t)BATCH * SEQ_LEN * ENC_IN;
    float* out_tro  = out_cr + (size_t)BATCH * SEQ_LEN * ENC_IN;

    // 1) RevIN normalize + transpose
    revin_norm_kernel<<<NROWS, 256, 0, stream>>>(x_in, rv_w, rv_b, z, mean, stdv);

    // 2) patch embed -> zt_a (token-major for the patch mamba)
    patch_embed_kernel<<<NROWS, 256, 0, stream>>>(z, ld1_w, ld1_b, zt_a);

    // 3) patch mamba x2 (layer 2 writes dense ld=63 so rows flatten to 4032)
    run_mamba(stream, m_patch, NROWS, zt_a, 64, zt_b, 64, 0, 0,
              xz, 252, xi, 128, xdbl, delta, 128, ybuf, 128);
    run_mamba(stream, m_patch, NROWS, zt_b, 64, ztf, 63, 0, 0,
              xz, 252, xi, 128, xdbl, delta, 128, ybuf, 128);

    // 4) zA = softmax(zt @ ltp_w + ltp_b)
    gemm(stream, ztf, 4032, ltp_w, K_TIME, zA, K_TIME, NROWS, K_TIME, 4032, ltp_b, 0, 0);
    softmax_rows_kernel<<<NROWS, 256, 0, stream>>>(zA, zA, K_TIME);

    // 5) time residual / forecast
    gemm(stream, zA, K_TIME, E_tp,  SEQ_LEN,  t_res,  SEQ_LEN,  NROWS, SEQ_LEN,  K_TIME, nullptr, 0, 0);
    gemm(stream, zA, K_TIME, E_tpf, PRED_LEN, z_time, PRED_LEN, NROWS, PRED_LEN, K_TIME, nullptr, 0, 0);

    // 6) channel branch: zc = z @ ldm_w + ldm_b
    gemm(stream, z, SEQ_LEN, ldm_w, D_MODEL, zc, D_MODEL, NROWS, D_MODEL, SEQ_LEN, ldm_b, 0, 0);

    // 7) bidirectional channel mamba (bwd fused via dir=1, accumulated)
    run_mamba(stream, m_chf, BATCH, zc, D_MODEL, zc2, D_MODEL, 0, 0,
              xz, 1024, xi, 512, xdbl, delta, 512, ybuf, 512);
    run_mamba(stream, m_chb, BATCH, zc, D_MODEL, zc2, D_MODEL, 1, 1,
              xz, 1024, xi, 512, xdbl, delta, 512, ybuf, 512);

    // 8) zcA = zc2 @ lch_w + lch_b
    gemm(stream, zc2, D_MODEL, lch_w, K_CH, zcA, K_CH, NROWS, K_CH, D_MODEL, lch_b, 0, 0);

    // 9) softmaxed channel dictionaries
    softmax_rows_kernel<<<K_CH, 256, 0, stream>>>(E_ch,  Esm,  SEQ_LEN);
    softmax_rows_kernel<<<K_CH, 256, 0, stream>>>(E_chf, EsmF, PRED_LEN);
    gemm(stream, zcA, K_CH, Esm,  SEQ_LEN,  ch_res, SEQ_LEN,  NROWS, SEQ_LEN,  K_CH, nullptr, 0, 0);
    gemm(stream, zcA, K_CH, EsmF, PRED_LEN, zch,    PRED_LEN, NROWS, PRED_LEN, K_CH, nullptr, 0, 0);

    // 10) z_out = [z_channel | z_time] @ wz_w + wz_b   (two accumulated GEMMs)
    gemm(stream, zch,    PRED_LEN, wz_w,                 PRED_LEN, zo, PRED_LEN, NROWS, PRED_LEN, PRED_LEN, wz_b, 0, 0);
    gemm(stream, z_time, PRED_LEN, wz_w + 96 * PRED_LEN, PRED_LEN, zo, PRED_LEN, NROWS, PRED_LEN, PRED_LEN, nullptr, 0, 1);

    // 11) tr = [time_res | ch_res] @ wr_w + wr_b
    gemm(stream, t_res,  SEQ_LEN, wr_w,                  SEQ_LEN, trb, SEQ_LEN, NROWS, SEQ_LEN, SEQ_LEN, wr_b, 0, 0);
    gemm(stream, ch_res, SEQ_LEN, wr_w + 512 * SEQ_LEN,  SEQ_LEN, trb, SEQ_LEN, NROWS, SEQ_LEN, SEQ_LEN, nullptr, 0, 1);

    // 12) denorm + transpose into the four outputs (return order)
    {
        const size_t n96  = (size_t)BATCH * PRED_LEN * ENC_IN;
        const size_t n512 = (size_t)BATCH * SEQ_LEN * ENC_IN;
        denorm_kernel<<<(int)((n96  + 255) / 256), 256, 0, stream>>>(zo,     PRED_LEN, rv_w, rv_b, mean, stdv, out_zout);
        denorm_kernel<<<(int)((n512 + 255) / 256), 256, 0, stream>>>(trb,    SEQ_LEN,  rv_w, rv_b, mean, stdv, out_tr);
        denorm_kernel<<<(int)((n512 + 255) / 256), 256, 0, stream>>>(ch_res, SEQ_LEN,  rv_w, rv_b, mean, stdv, out_cr);
        denorm_kernel<<<(int)((n512 + 255) / 256), 256, 0, stream>>>(t_res,  SEQ_LEN,  rv_w, rv_b, mean, stdv, out_tro);
    }
}